// HiddenFusion_44074954392050
// MI455X (gfx1250) — compile-verified
//
#include <hip/hip_runtime.h>
#include <hip/hip_bf16.h>

// ---------------- constants (from reference) ----------------
#define L_SEQ   512
#define BATCH   16
#define D_MODEL 512
#define D_STATE 64
#define D_INNER 1024      // 2*D_MODEL
#define DT_RANK 32
#define D_CONV  4
#define ROWS    (L_SEQ * BATCH)          // 8192 rows, row r = l*BATCH + b

typedef __attribute__((ext_vector_type(2))) float v2f;
typedef __attribute__((ext_vector_type(8))) float v8f;

#define WMMA_F32(A_, B_, C_) \
    __builtin_amdgcn_wmma_f32_16x16x4_f32(false, (A_), false, (B_), (short)0, (C_), false, false)

// ---------------- fp32 WMMA GEMM:  C = A(ROWS x K, lda) * Bw(N x K, ldb)^T ----------------
// block = 256 threads = 8 waves, 2(M) x 4(N); wave tile 32 x 64; block tile 64 x 256.
// Software-pipelined: fragments for step k+4 are loaded before the WMMAs of step k.
// All N used (2048/1024/512/160) are multiples of 16 -> 16-col group validity is
// wave-uniform; invalid groups clamp to row 0 and their stores are skipped.
__global__ __launch_bounds__(256) void wmma_gemm_f32(
    const float* __restrict__ A, int lda,
    const float* __restrict__ Bw, int ldb,
    float* __restrict__ C, int ldc,
    int N, int K)
{
    const int lane = threadIdx.x & 31;
    const int wave = threadIdx.x >> 5;
    const int wm   = wave >> 2;          // 0..1
    const int wn   = wave & 3;           // 0..3
    const int lm   = lane & 15;          // row-within-tile
    const int lk   = lane >> 4;          // 0/1 -> K pair select

    const int m0    = blockIdx.y * 64 + wm * 32;
    const int nBase = blockIdx.x * 256 + wn * 64;

    // wave-uniform group validity (scalar)
    const bool v0 = (nBase +  0) < N;
    const bool v1 = (nBase + 16) < N;
    const bool v2 = (nBase + 32) < N;
    const bool v3 = (nBase + 48) < N;

    // 32-bit lane offsets against uniform base pointers
    const int aofs0 = (m0 +      lm) * lda + 2 * lk;
    const int aofs1 = (m0 + 16 + lm) * lda + 2 * lk;
    const int bofs0 = ((v0 ? nBase +  0 : 0) + lm) * ldb + 2 * lk;
    const int bofs1 = ((v1 ? nBase + 16 : 0) + lm) * ldb + 2 * lk;
    const int bofs2 = ((v2 ? nBase + 32 : 0) + lm) * ldb + 2 * lk;
    const int bofs3 = ((v3 ? nBase + 48 : 0) + lm) * ldb + 2 * lk;

    v8f acc00 = {}, acc01 = {}, acc02 = {}, acc03 = {};
    v8f acc10 = {}, acc11 = {}, acc12 = {}, acc13 = {};

    // pipeline prologue
    v2f a0 = *(const v2f*)(A  + aofs0);
    v2f a1 = *(const v2f*)(A  + aofs1);
    v2f b0 = *(const v2f*)(Bw + bofs0);
    v2f b1 = *(const v2f*)(Bw + bofs1);
    v2f b2 = *(const v2f*)(Bw + bofs2);
    v2f b3 = *(const v2f*)(Bw + bofs3);

    #pragma unroll 2
    for (int k0 = 4; k0 < K; k0 += 4) {
        v2f a0n = *(const v2f*)(A  + aofs0 + k0);
        v2f a1n = *(const v2f*)(A  + aofs1 + k0);
        v2f b0n = *(const v2f*)(Bw + bofs0 + k0);
        v2f b1n = *(const v2f*)(Bw + bofs1 + k0);
        v2f b2n = *(const v2f*)(Bw + bofs2 + k0);
        v2f b3n = *(const v2f*)(Bw + bofs3 + k0);

        acc00 = WMMA_F32(a0, b0, acc00);
        acc01 = WMMA_F32(a0, b1, acc01);
        acc02 = WMMA_F32(a0, b2, acc02);
        acc03 = WMMA_F32(a0, b3, acc03);
        acc10 = WMMA_F32(a1, b0, acc10);
        acc11 = WMMA_F32(a1, b1, acc11);
        acc12 = WMMA_F32(a1, b2, acc12);
        acc13 = WMMA_F32(a1, b3, acc13);

        a0 = a0n; a1 = a1n;
        b0 = b0n; b1 = b1n; b2 = b2n; b3 = b3n;
    }

    // pipeline epilogue
    acc00 = WMMA_F32(a0, b0, acc00);
    acc01 = WMMA_F32(a0, b1, acc01);
    acc02 = WMMA_F32(a0, b2, acc02);
    acc03 = WMMA_F32(a0, b3, acc03);
    acc10 = WMMA_F32(a1, b0, acc10);
    acc11 = WMMA_F32(a1, b1, acc11);
    acc12 = WMMA_F32(a1, b2, acc12);
    acc13 = WMMA_F32(a1, b3, acc13);

    // D layout: VGPR r -> M = mtile + r + 8*lk, N = nBase + j*16 + lm
    const int n0 = nBase + lm;
    #pragma unroll
    for (int r = 0; r < 8; ++r) {
        float* crow0 = C + (size_t)(m0 +      r + 8 * lk) * ldc;
        float* crow1 = C + (size_t)(m0 + 16 + r + 8 * lk) * ldc;
        if (v0) { crow0[n0     ] = acc00[r]; crow1[n0     ] = acc10[r]; }
        if (v1) { crow0[n0 + 16] = acc01[r]; crow1[n0 + 16] = acc11[r]; }
        if (v2) { crow0[n0 + 32] = acc02[r]; crow1[n0 + 32] = acc12[r]; }
        if (v3) { crow0[n0 + 48] = acc03[r]; crow1[n0 + 48] = acc13[r]; }
    }
}

// ---------------- depthwise causal conv(4) + SiLU ----------------
// xz: (ROWS, 2048) -> xc is cols [0,1024);  xs out: (ROWS, 1024)
__global__ __launch_bounds__(256) void conv_silu_kernel(
    const float* __restrict__ xz, const float* __restrict__ cw,
    const float* __restrict__ cb, float* __restrict__ xs)
{
    int i = blockIdx.x * blockDim.x + threadIdx.x;
    if (i >= ROWS * D_INNER) return;
    int d = i & (D_INNER - 1);
    int r = i >> 10;
    float acc = cb[d];
    const float* wd = cw + d * D_CONV;
    #pragma unroll
    for (int k = 0; k < D_CONV; ++k) {
        int rp = r - (D_CONV - 1 - k) * BATCH;   // same b, time t-3+k
        if (rp >= 0) acc = fmaf(xz[(size_t)rp * (2 * D_INNER) + d], wd[k], acc);
    }
    xs[i] = acc / (1.0f + __expf(-acc));
}

// ---------------- helpers ----------------
__device__ __forceinline__ float softplus_f(float x) {
    float e = __expf(-fabsf(x));
    float r = __logf(1.0f + e);
    return x > 0.0f ? x + r : r;
}
__device__ __forceinline__ float silu_f(float x) {
    return x / (1.0f + __expf(-x));
}

// ---------------- coupled SSM scan ----------------
// one wave per (b, d); 64 states as 2 regs/lane (n = lane, lane+32)
// grid = (D_INNER/8, BATCH), block = 256 (8 waves -> 8 channels)
__global__ __launch_bounds__(256) void mamba_scan_kernel(
    const float* xsA, const float* __restrict__ dtpA,
    const float* __restrict__ xdbA, const float* __restrict__ xzA,
    const float* __restrict__ AlogA, const float* __restrict__ dtbA,
    const float* __restrict__ DDA,
    const float* xsV, const float* __restrict__ dtpV,
    const float* __restrict__ xdbV, const float* __restrict__ xzV,
    const float* __restrict__ AlogV, const float* __restrict__ dtbV,
    const float* __restrict__ DDV,
    float* yA, float* yV)          // yA aliases xsA, yV aliases xsV (in-place)
{
    const int b    = blockIdx.y;
    const int wave = threadIdx.x >> 5;
    const int lane = threadIdx.x & 31;
    const int d    = blockIdx.x * 8 + wave;

    const float Aa0 = -__expf(AlogA[d * D_STATE + lane]);
    const float Aa1 = -__expf(AlogA[d * D_STATE + lane + 32]);
    const float Av0 = -__expf(AlogV[d * D_STATE + lane]);
    const float Av1 = -__expf(AlogV[d * D_STATE + lane + 32]);
    const float dtba = dtbA[d], dtbv = dtbV[d];
    const float Da = DDA[d],   Dv = DDV[d];

    float s0 = 0.0f, s1 = 0.0f;      // coupled state s = ssa + ssv

    for (int t = 0; t < L_SEQ; ++t) {
        const size_t row = (size_t)t * BATCH + b;

        const float* xa_p = xdbA + row * 160;
        const float* xv_p = xdbV + row * 160;
        float Ba0 = xa_p[32 + lane],  Ba1 = xa_p[64 + lane];
        float Ca0 = xa_p[96 + lane],  Ca1 = xa_p[128 + lane];
        float Bv0 = xv_p[32 + lane],  Bv1 = xv_p[64 + lane];
        float Cv0 = xv_p[96 + lane],  Cv1 = xv_p[128 + lane];

        float dta = softplus_f(dtpA[row * D_INNER + d] + dtba);
        float dtv = softplus_f(dtpV[row * D_INNER + d] + dtbv);
        float xa  = xsA[row * D_INNER + d];
        float xv  = xsV[row * D_INNER + d];
        float dtxa = dta * xa;
        float dtxv = dtv * xv;

        float ea0 = __expf(dta * Aa0), ea1 = __expf(dta * Aa1);
        float ev0 = __expf(dtv * Av0), ev1 = __expf(dtv * Av1);

        float sa0 = fmaf(s0, ea0, dtxa * Ba0);
        float sa1 = fmaf(s1, ea1, dtxa * Ba1);
        float sv0 = fmaf(s0, ev0, dtxv * Bv0);
        float sv1 = fmaf(s1, ev1, dtxv * Bv1);
        s0 = sa0 + sv0;
        s1 = sa1 + sv1;

        float ya = fmaf(sa0, Ca0, sa1 * Ca1);
        float yv = fmaf(sv0, Cv0, sv1 * Cv1);
        #pragma unroll
        for (int off = 16; off > 0; off >>= 1) {
            ya += __shfl_xor(ya, off, 32);
            yv += __shfl_xor(yv, off, 32);
        }

        if (lane == 0) {
            float za = xzA[row * (2 * D_INNER) + D_INNER + d];
            float zv = xzV[row * (2 * D_INNER) + D_INNER + d];
            yA[row * D_INNER + d] = (ya + Da * xa) * silu_f(za);
            yV[row * D_INNER + d] = (yv + Dv * xv) * silu_f(zv);
        }
    }
}

// ---------------- host side ----------------
static inline void launch_gemm(const float* A, int lda, const float* Bw, int ldb,
                               float* C, int ldc, int N, int K, hipStream_t s)
{
    dim3 grid((N + 255) / 256, ROWS / 64);
    wmma_gemm_f32<<<grid, 256, 0, s>>>(A, lda, Bw, ldb, C, ldc, N, K);
}

extern "C" void kernel_launch(void* const* d_in, const int* in_sizes, int n_in,
                              void* d_out, int out_size, void* d_ws, size_t ws_size,
                              hipStream_t stream)
{
    const float* a = (const float*)d_in[0];
    const float* v = (const float*)d_in[1];
    // params_a: W_in, conv_w, conv_b, W_x, W_dt, dt_b, A_log, D, W_out
    const float* WinA  = (const float*)d_in[2];
    const float* cwA   = (const float*)d_in[3];
    const float* cbA   = (const float*)d_in[4];
    const float* WxA   = (const float*)d_in[5];
    const float* WdtA  = (const float*)d_in[6];
    const float* dtbA  = (const float*)d_in[7];
    const float* AlogA = (const float*)d_in[8];
    const float* DDA   = (const float*)d_in[9];
    const float* WoutA = (const float*)d_in[10];
    const float* WinV  = (const float*)d_in[11];
    const float* cwV   = (const float*)d_in[12];
    const float* cbV   = (const float*)d_in[13];
    const float* WxV   = (const float*)d_in[14];
    const float* WdtV  = (const float*)d_in[15];
    const float* dtbV  = (const float*)d_in[16];
    const float* AlogV = (const float*)d_in[17];
    const float* DDV   = (const float*)d_in[18];
    const float* WoutV = (const float*)d_in[19];

    // workspace layout (floats)
    const size_t XZ  = (size_t)ROWS * 2 * D_INNER;   // 16,777,216
    const size_t XS  = (size_t)ROWS * D_INNER;       //  8,388,608
    const size_t XDB = (size_t)ROWS * 160;           //  1,310,720
    float* ws   = (float*)d_ws;
    float* xzA  = ws;
    float* xzV  = xzA + XZ;
    float* xsA  = xzV + XZ;
    float* xsV  = xsA + XS;
    float* xdbA = xsV + XS;
    float* xdbV = xdbA + XDB;
    float* dtpA = xdbV + XDB;
    float* dtpV = dtpA + XS;

    float* outA = (float*)d_out;
    float* outV = outA + (size_t)ROWS * D_MODEL;

    // 1) input projection: xz = X @ W_in^T   (N=2048, K=512)
    launch_gemm(a, D_MODEL, WinA, D_MODEL, xzA, 2 * D_INNER, 2 * D_INNER, D_MODEL, stream);
    launch_gemm(v, D_MODEL, WinV, D_MODEL, xzV, 2 * D_INNER, 2 * D_INNER, D_MODEL, stream);

    // 2) causal depthwise conv + SiLU -> xs
    int ethreads = ROWS * D_INNER;
    conv_silu_kernel<<<(ethreads + 255) / 256, 256, 0, stream>>>(xzA, cwA, cbA, xsA);
    conv_silu_kernel<<<(ethreads + 255) / 256, 256, 0, stream>>>(xzV, cwV, cbV, xsV);

    // 3) xdb = xs @ W_x^T   (N=160, K=1024)
    launch_gemm(xsA, D_INNER, WxA, D_INNER, xdbA, 160, 160, D_INNER, stream);
    launch_gemm(xsV, D_INNER, WxV, D_INNER, xdbV, 160, 160, D_INNER, stream);

    // 4) dt_pre = xdb[:, :32] @ W_dt^T   (N=1024, K=32)
    launch_gemm(xdbA, 160, WdtA, DT_RANK, dtpA, D_INNER, D_INNER, DT_RANK, stream);
    launch_gemm(xdbV, 160, WdtV, DT_RANK, dtpV, D_INNER, D_INNER, DT_RANK, stream);

    // 5) coupled scan (softplus + gating fused); writes y in place of xs
    mamba_scan_kernel<<<dim3(D_INNER / 8, BATCH), 256, 0, stream>>>(
        xsA, dtpA, xdbA, xzA, AlogA, dtbA, DDA,
        xsV, dtpV, xdbV, xzV, AlogV, dtbV, DDV,
        xsA, xsV);

    // 6) out = y @ W_out^T   (N=512, K=1024)
    launch_gemm(xsA, D_INNER, WoutA, D_INNER, outA, D_MODEL, D_MODEL, D_INNER, stream);
    launch_gemm(xsV, D_INNER, WoutV, D_INNER, outV, D_MODEL, D_MODEL, D_INNER, stream);
}